// GLASS_54932631716167
// MI455X (gfx1250) — compile-verified
//
#include <hip/hip_runtime.h>
#include <math.h>

#define N_NODES 40000
#define N_EDGES 640000
#define DIN     127
#define H       128
#define NSEG    512
#define SEG     64

typedef __attribute__((ext_vector_type(2))) float v2f;
typedef __attribute__((ext_vector_type(8))) float v8f;

// ---------------------------------------------------------------------------
// Build h0 = [x | batch-indicator column]
// ---------------------------------------------------------------------------
__global__ void build_h0(const float* __restrict__ x, float* __restrict__ h0) {
    int gid = blockIdx.x * blockDim.x + threadIdx.x;   // N*H threads
    if (gid >= N_NODES * H) return;
    int n = gid >> 7;
    int c = gid & (H - 1);
    h0[gid] = (c < DIN) ? x[n * DIN + c] : 0.0f;
}

__global__ void scatter_bf(const int* __restrict__ batches, float* __restrict__ h0) {
    int i = blockIdx.x * blockDim.x + threadIdx.x;
    if (i >= NSEG * SEG) return;
    h0[batches[i] * H + (H - 1)] = 1.0f;   // duplicates all write 1.0 -> fine
}

// ---------------------------------------------------------------------------
// agg init: agg = h   (then edges atomically accumulate on top -> h + agg)
// ---------------------------------------------------------------------------
__global__ void copy_f4(const float4* __restrict__ src, float4* __restrict__ dst, int n4) {
    int i = blockIdx.x * blockDim.x + threadIdx.x;
    if (i < n4) dst[i] = src[i];
}

// ---------------------------------------------------------------------------
// Edge aggregation: agg[dst] += ew * h[src]   (float4 per thread, fp32 atomics)
// ---------------------------------------------------------------------------
__global__ void edge_agg(const int* __restrict__ esrc, const int* __restrict__ edst,
                         const float* __restrict__ ew, const float* __restrict__ h,
                         float* __restrict__ agg) {
    int gid = blockIdx.x * blockDim.x + threadIdx.x;   // E * 32 threads
    int e  = gid >> 5;
    if (e >= N_EDGES) return;
    int cb = (gid & 31) << 2;                          // channel base: 0,4,...,124
    float w = ew[e];
    const float4 v = *reinterpret_cast<const float4*>(h + esrc[e] * H + cb);
    float* p = agg + edst[e] * H + cb;
    unsafeAtomicAdd(p + 0, w * v.x);
    unsafeAtomicAdd(p + 1, w * v.y);
    unsafeAtomicAdd(p + 2, w * v.z);
    unsafeAtomicAdd(p + 3, w * v.w);
}

// ---------------------------------------------------------------------------
// out = relu(A @ W + bias), A:[N,128] W:[128,128], fp32 WMMA 16x16x4
// One block = 8 waves = 16 rows x 128 cols; wave w owns column tile w.
// ---------------------------------------------------------------------------
__global__ __launch_bounds__(256) void gin_gemm_relu(
    const float* __restrict__ A, const float* __restrict__ W,
    const float* __restrict__ bias, float* __restrict__ out) {
    const int lane = threadIdx.x & 31;
    const int wave = threadIdx.x >> 5;          // 0..7 -> column tile
    const int m0   = blockIdx.x << 4;           // row base (N divisible by 16)
    const int l15  = lane & 15;
    const int col  = l15 + (wave << 4);
    const int khalf = (lane >> 4) << 1;         // lanes 16..31 hold K=2,3
    const float* arow = A + (m0 + l15) * H;

    v8f acc = {};
#pragma unroll
    for (int k = 0; k < H; k += 4) {
        v2f a, b;
        a.x = arow[k + khalf];
        a.y = arow[k + khalf + 1];
        b.x = W[(k + khalf) * H + col];
        b.y = W[(k + khalf + 1) * H + col];
        acc = __builtin_amdgcn_wmma_f32_16x16x4_f32(
            /*neg_a=*/false, a, /*neg_b=*/false, b,
            /*c_mod=*/(short)0, acc, /*reuse_a=*/false, /*reuse_b=*/false);
    }

    const int rbase = m0 + ((lane >> 4) << 3);  // M split at lane 16
    const float bv = bias[col];
#pragma unroll
    for (int j = 0; j < 8; ++j) {
        float v = acc[j] + bv;
        out[(rbase + j) * H + col] = v > 0.0f ? v : 0.0f;
    }
}

// ---------------------------------------------------------------------------
// MultiPool: per segment (64 gathered rows) mean / min / max -> z[NSEG, 3H]
// ---------------------------------------------------------------------------
__global__ void multipool(const float* __restrict__ h, const int* __restrict__ batches,
                          float* __restrict__ z) {
    int s = blockIdx.x;       // 0..NSEG-1
    int c = threadIdx.x;      // 0..H-1
    const int* bi = batches + s * SEG;
    float sum = 0.0f, mn = 3.4e38f, mx = -3.4e38f;
#pragma unroll 4
    for (int i = 0; i < SEG; ++i) {
        float v = h[bi[i] * H + c];
        sum += v;
        mn = fminf(mn, v);
        mx = fmaxf(mx, v);
    }
    float* zr = z + s * (3 * H);
    zr[c]         = sum * (1.0f / (float)SEG);
    zr[H + c]     = mn;
    zr[2 * H + c] = mx;
}

// ---------------------------------------------------------------------------
// Block-wide sum over H=128 threads (LDS tree)
// ---------------------------------------------------------------------------
__device__ __forceinline__ float block_sum(float v, float* sh, int c) {
    sh[c] = v;
    __syncthreads();
    for (int off = H >> 1; off > 0; off >>= 1) {
        if (c < off) sh[c] += sh[c + off];
        __syncthreads();
    }
    float r = sh[0];
    __syncthreads();
    return r;
}

// t1 = LN(relu(z @ w1 + b1))
__global__ void readout1(const float* __restrict__ z, const float* __restrict__ w1,
                         const float* __restrict__ b1, const float* __restrict__ g,
                         const float* __restrict__ be, float* __restrict__ t1) {
    __shared__ float sh[H];
    int s = blockIdx.x, c = threadIdx.x;
    const float* zr = z + s * (3 * H);
    float acc = b1[c];
    for (int k = 0; k < 3 * H; ++k) acc += zr[k] * w1[k * H + c];
    acc = fmaxf(acc, 0.0f);
    float mean = block_sum(acc, sh, c) * (1.0f / (float)H);
    float d = acc - mean;
    float var = block_sum(d * d, sh, c) * (1.0f / (float)H);
    t1[s * H + c] = d * rsqrtf(var + 1e-5f) * g[c] + be[c];
}

// t2 = relu(LN(t1 @ w2 + b2))
__global__ void readout2(const float* __restrict__ t1, const float* __restrict__ w2,
                         const float* __restrict__ b2, const float* __restrict__ g,
                         const float* __restrict__ be, float* __restrict__ t2) {
    __shared__ float sh[H];
    int s = blockIdx.x, c = threadIdx.x;
    const float* tr = t1 + s * H;
    float acc = b2[c];
    for (int k = 0; k < H; ++k) acc += tr[k] * w2[k * H + c];
    float mean = block_sum(acc, sh, c) * (1.0f / (float)H);
    float d = acc - mean;
    float var = block_sum(d * d, sh, c) * (1.0f / (float)H);
    float v = d * rsqrtf(var + 1e-5f) * g[c] + be[c];
    t2[s * H + c] = fmaxf(v, 0.0f);
}

// out = t2 @ w3 + b3   ([NSEG,1])
__global__ void readout3(const float* __restrict__ t2, const float* __restrict__ w3,
                         const float* __restrict__ b3, float* __restrict__ out) {
    __shared__ float sh[H];
    int s = blockIdx.x, c = threadIdx.x;
    float v = t2[s * H + c] * w3[c];
    float sum = block_sum(v, sh, c);
    if (c == 0) out[s] = sum + b3[0];
}

// ---------------------------------------------------------------------------
extern "C" void kernel_launch(void* const* d_in, const int* in_sizes, int n_in,
                              void* d_out, int out_size, void* d_ws, size_t ws_size,
                              hipStream_t stream) {
    const float* x       = (const float*)d_in[0];
    const int*   ei      = (const int*)  d_in[1];   // [2,E]
    const float* ew      = (const float*)d_in[2];
    const int*   batches = (const int*)  d_in[3];
    // d_in[4] = lens (all segments length 64, unused)
    const float* gw[3] = { (const float*)d_in[5], (const float*)d_in[7], (const float*)d_in[9]  };
    const float* gb[3] = { (const float*)d_in[6], (const float*)d_in[8], (const float*)d_in[10] };
    const float* r_w1  = (const float*)d_in[11];
    const float* r_b1  = (const float*)d_in[12];
    const float* ln1_g = (const float*)d_in[13];
    const float* ln1_b = (const float*)d_in[14];
    const float* r_w2  = (const float*)d_in[15];
    const float* r_b2  = (const float*)d_in[16];
    const float* ln2_g = (const float*)d_in[17];
    const float* ln2_b = (const float*)d_in[18];
    const float* r_w3  = (const float*)d_in[19];
    const float* r_b3  = (const float*)d_in[20];
    float* out = (float*)d_out;

    const int* esrc = ei;
    const int* edst = ei + N_EDGES;

    // workspace carve-up
    float* bufA = (float*)d_ws;                    // [N,H] current h
    float* bufB = bufA + (size_t)N_NODES * H;      // [N,H] agg = h + edge sums
    float* z    = bufB + (size_t)N_NODES * H;      // [NSEG, 3H]
    float* t1   = z    + (size_t)NSEG * 3 * H;     // [NSEG, H]
    float* t2   = t1   + (size_t)NSEG * H;         // [NSEG, H]

    const int NH  = N_NODES * H;                   // 5,120,000
    const int NH4 = NH / 4;

    // h0 = [x | batch flag]
    build_h0<<<(NH + 255) / 256, 256, 0, stream>>>(x, bufA);
    scatter_bf<<<(NSEG * SEG + 255) / 256, 256, 0, stream>>>(batches, bufA);

    // 3 GIN layers: agg = h; agg += edges; h = relu(agg @ W + b)
    for (int l = 0; l < 3; ++l) {
        copy_f4<<<(NH4 + 255) / 256, 256, 0, stream>>>(
            (const float4*)bufA, (float4*)bufB, NH4);
        edge_agg<<<(N_EDGES * 32 + 255) / 256, 256, 0, stream>>>(
            esrc, edst, ew, bufA, bufB);
        gin_gemm_relu<<<N_NODES / 16, 256, 0, stream>>>(bufB, gw[l], gb[l], bufA);
    }

    // pool + readout
    multipool<<<NSEG, H, 0, stream>>>(bufA, batches, z);
    readout1<<<NSEG, H, 0, stream>>>(z, r_w1, r_b1, ln1_g, ln1_b, t1);
    readout2<<<NSEG, H, 0, stream>>>(t1, r_w2, r_b2, ln2_g, ln2_b, t2);
    readout3<<<NSEG, H, 0, stream>>>(t2, r_w3, r_b3, out);
}